// DifferentiableAddressingHead_84138409328920
// MI455X (gfx1250) — compile-verified
//
#include <hip/hip_runtime.h>
#include <math.h>

typedef float v2f __attribute__((ext_vector_type(2)));
typedef float v8f __attribute__((ext_vector_type(8)));
typedef float f4  __attribute__((ext_vector_type(4)));

#define NTM_B 64
#define NTM_N 8192
#define NTM_D 128
#define NTM_C 256
#define NTM_EPS 1e-8f

// ---------------------------------------------------------------------------
// Kernel 1: key = controller_state @ Wk   (64x256 @ 256x128, fp32)
// One wave (32 threads) per 16x16 output tile, V_WMMA_F32_16X16X4_F32,
// 64 K-steps of 4. Lane layout per CDNA5 ISA 7.12.2:
//   A 16x4:  lanes 0-15 hold {K=k0,k0+1} of row M=lane; lanes 16-31 hold {k0+2,k0+3}
//   B 4x16:  lanes 0-15 hold rows K=k0,k0+1 at N=lane; lanes 16-31 rows k0+2,k0+3
//   C/D 16x16: VGPR i -> M = i (+8 for upper lane half), N = lane&15
// ---------------------------------------------------------------------------
__global__ void ntm_key_gemm(const float* __restrict__ cs,
                             const float* __restrict__ Wk,
                             float* __restrict__ key) {
  const int n0   = blockIdx.x * 16;     // column tile in D (0..7)
  const int m0   = blockIdx.y * 16;     // row tile in B (0..3)
  const int lane = threadIdx.x;
  const int half = lane >> 4;           // 0 or 1
  const int r    = lane & 15;

#if __has_builtin(__builtin_amdgcn_wmma_f32_16x16x4_f32)
  v8f acc = {};
  const float* arow = cs + (size_t)(m0 + r) * NTM_C + 2 * half;
  const float* bcol = Wk + (size_t)(2 * half) * NTM_D + n0 + r;
  for (int k0 = 0; k0 < NTM_C; k0 += 4) {
    v2f a, b;
    a.x = arow[k0];
    a.y = arow[k0 + 1];
    b.x = bcol[(size_t)k0 * NTM_D];
    b.y = bcol[(size_t)(k0 + 1) * NTM_D];
    acc = __builtin_amdgcn_wmma_f32_16x16x4_f32(
        /*neg_a=*/false, a, /*neg_b=*/false, b,
        /*c_mod=*/(short)0, acc, /*reuse_a=*/false, /*reuse_b=*/false);
  }
#pragma unroll
  for (int i = 0; i < 8; ++i)
    key[(size_t)(m0 + half * 8 + i) * NTM_D + n0 + r] = acc[i];
#else
  // Scalar fallback (should not be taken on gfx1250).
  for (int i = 0; i < 8; ++i) {
    const int row = m0 + half * 8 + i;
    const int col = n0 + r;
    float s = 0.f;
    for (int k = 0; k < NTM_C; ++k)
      s += cs[(size_t)row * NTM_C + k] * Wk[(size_t)k * NTM_D + col];
    key[(size_t)row * NTM_D + col] = s;
  }
#endif
}

// ---------------------------------------------------------------------------
// Kernel 2: per-batch scalar params.
// params[b*8 + {0:beta, 1:gate, 2:gamma, 3:key_norm, 4..6:shift softmax}]
// ---------------------------------------------------------------------------
__device__ __forceinline__ float ntm_softplus(float x) {
  return (x > 20.f) ? x : log1pf(expf(x));
}

__global__ void ntm_params(const float* __restrict__ cs,
                           const float* __restrict__ key,
                           const float* __restrict__ Wb,  const float* __restrict__ bb,
                           const float* __restrict__ Wg,  const float* __restrict__ bg,
                           const float* __restrict__ Ws,  const float* __restrict__ bs,
                           const float* __restrict__ Wgam,const float* __restrict__ bgam,
                           float* __restrict__ params) {
  __shared__ float sd[7 * 256];
  const int b = blockIdx.x;
  const int t = threadIdx.x;

  const float c = cs[(size_t)b * NTM_C + t];
  sd[0 * 256 + t] = c * Wb[t];
  sd[1 * 256 + t] = c * Wg[t];
  sd[2 * 256 + t] = c * Wgam[t];
  sd[3 * 256 + t] = c * Ws[t * 3 + 0];
  sd[4 * 256 + t] = c * Ws[t * 3 + 1];
  sd[5 * 256 + t] = c * Ws[t * 3 + 2];
  const float kv = (t < NTM_D) ? key[(size_t)b * NTM_D + t] : 0.f;
  sd[6 * 256 + t] = kv * kv;
  __syncthreads();

  for (int s = 128; s > 0; s >>= 1) {
    if (t < s) {
#pragma unroll
      for (int j = 0; j < 7; ++j) sd[j * 256 + t] += sd[j * 256 + t + s];
    }
    __syncthreads();
  }

  if (t == 0) {
    const float beta  = ntm_softplus(sd[0 * 256] + bb[0]) + 1.0f;
    const float gate  = 1.0f / (1.0f + expf(-(sd[1 * 256] + bg[0])));
    const float gamma = ntm_softplus(sd[2 * 256] + bgam[0]) + 1.0f;
    const float knorm = sqrtf(sd[6 * 256]);
    float x0 = sd[3 * 256] + bs[0];
    float x1 = sd[4 * 256] + bs[1];
    float x2 = sd[5 * 256] + bs[2];
    const float m = fmaxf(x0, fmaxf(x1, x2));
    const float e0 = expf(x0 - m), e1 = expf(x1 - m), e2 = expf(x2 - m);
    const float inv = 1.0f / (e0 + e1 + e2);
    float* p = params + b * 8;
    p[0] = beta;  p[1] = gate;  p[2] = gamma; p[3] = knorm;
    p[4] = e0 * inv; p[5] = e1 * inv; p[6] = e2 * inv;
  }
}

// ---------------------------------------------------------------------------
// Kernel 3: the 256 MB streaming pass.
// One wave per row: 32 lanes x float4 = one full 512 B row (coalesced).
// Fused: dot(mem,key), ||mem||^2, z = beta * dot / (knorm*||mem|| + eps).
// Non-temporal loads: memory (256 MB) > L2 (192 MB), read exactly once.
// ---------------------------------------------------------------------------
__global__ void ntm_dot(const float* __restrict__ memory,
                        const float* __restrict__ key,
                        const float* __restrict__ params,
                        float* __restrict__ z) {
  const int b    = blockIdx.y;
  const int w    = threadIdx.x >> 5;
  const int lane = threadIdx.x & 31;

  const f4 k4 = ((const f4*)(key + (size_t)b * NTM_D))[lane];
  const float beta  = params[b * 8 + 0];
  const float knorm = params[b * 8 + 3];

  const int base = blockIdx.x * 64 + w * 8;  // 8 waves/block, 8 rows/wave
#pragma unroll
  for (int r = 0; r < 8; ++r) {
    const int n = base + r;
    const f4* mp = (const f4*)(memory + ((size_t)b * NTM_N + n) * NTM_D);
    const f4 m4 = __builtin_nontemporal_load(mp + lane);
    float pd = m4.x * k4.x + m4.y * k4.y + m4.z * k4.z + m4.w * k4.w;
    float ps = m4.x * m4.x + m4.y * m4.y + m4.z * m4.z + m4.w * m4.w;
#pragma unroll
    for (int off = 16; off > 0; off >>= 1) {
      pd += __shfl_down(pd, off);
      ps += __shfl_down(ps, off);
    }
    if (lane == 0)
      z[(size_t)b * NTM_N + n] = beta * pd / (knorm * sqrtf(ps) + NTM_EPS);
  }
}

// ---------------------------------------------------------------------------
// Kernel 4: per-batch row ops. One 1024-thread block per batch; row in LDS.
// softmax -> gate blend -> circular 3-tap conv -> pow sharpen -> normalize.
// ---------------------------------------------------------------------------
__device__ __forceinline__ float ntm_block_reduce_sum(float v, float* red) {
#pragma unroll
  for (int off = 16; off > 0; off >>= 1) v += __shfl_down(v, off);
  const int wid = threadIdx.x >> 5, lane = threadIdx.x & 31;
  if (lane == 0) red[wid] = v;
  __syncthreads();
  if (threadIdx.x < 32) {
    float x = red[threadIdx.x];
#pragma unroll
    for (int off = 16; off > 0; off >>= 1) x += __shfl_down(x, off);
    if (threadIdx.x == 0) red[0] = x;
  }
  __syncthreads();
  const float r = red[0];
  __syncthreads();
  return r;
}

__device__ __forceinline__ float ntm_block_reduce_max(float v, float* red) {
#pragma unroll
  for (int off = 16; off > 0; off >>= 1) v = fmaxf(v, __shfl_down(v, off));
  const int wid = threadIdx.x >> 5, lane = threadIdx.x & 31;
  if (lane == 0) red[wid] = v;
  __syncthreads();
  if (threadIdx.x < 32) {
    float x = red[threadIdx.x];
#pragma unroll
    for (int off = 16; off > 0; off >>= 1) x = fmaxf(x, __shfl_down(x, off));
    if (threadIdx.x == 0) red[0] = x;
  }
  __syncthreads();
  const float r = red[0];
  __syncthreads();
  return r;
}

__global__ void ntm_rowops(const float* __restrict__ z,
                           const float* __restrict__ prev,
                           const float* __restrict__ params,
                           float* __restrict__ out) {
  __shared__ float row[NTM_N];   // 32 KB of 320 KB LDS
  __shared__ float red[32];

  const int b   = blockIdx.x;
  const int tid = threadIdx.x;
  const float gate  = params[b * 8 + 1];
  const float gamma = params[b * 8 + 2];
  const float s0 = params[b * 8 + 4];
  const float s1 = params[b * 8 + 5];
  const float s2 = params[b * 8 + 6];

  float zv[8];
  float lmax = -3.4e38f;
#pragma unroll
  for (int i = 0; i < 8; ++i) {
    const int n = tid + i * 1024;
    zv[i] = z[(size_t)b * NTM_N + n];
    lmax = fmaxf(lmax, zv[i]);
  }
  const float rmax = ntm_block_reduce_max(lmax, red);

  float lsum = 0.f;
#pragma unroll
  for (int i = 0; i < 8; ++i) {
    zv[i] = expf(zv[i] - rmax);
    lsum += zv[i];
  }
  const float invsum = 1.0f / ntm_block_reduce_sum(lsum, red);

#pragma unroll
  for (int i = 0; i < 8; ++i) {
    const int n = tid + i * 1024;
    const float cw = zv[i] * invsum;
    row[n] = gate * cw + (1.0f - gate) * prev[(size_t)b * NTM_N + n];
  }
  __syncthreads();

  // shifted[n] = s0*w[(n+1)%N] + s1*w[n] + s2*w[(n-1)%N]   (offsets -1,0,+1)
  float sh[8];
  float lsh = 0.f;
#pragma unroll
  for (int i = 0; i < 8; ++i) {
    const int n = tid + i * 1024;
    const float right = row[(n + 1) & (NTM_N - 1)];
    const float mid   = row[n];
    const float left  = row[(n + NTM_N - 1) & (NTM_N - 1)];
    const float shifted = s0 * right + s1 * mid + s2 * left;
    sh[i] = powf(shifted + 1e-8f, gamma);
    lsh += sh[i];
  }
  const float invt = 1.0f / (ntm_block_reduce_sum(lsh, red) + NTM_EPS);

#pragma unroll
  for (int i = 0; i < 8; ++i) {
    const int n = tid + i * 1024;
    out[(size_t)b * NTM_N + n] = sh[i] * invt;
  }
}

// ---------------------------------------------------------------------------
extern "C" void kernel_launch(void* const* d_in, const int* in_sizes, int n_in,
                              void* d_out, int out_size, void* d_ws, size_t ws_size,
                              hipStream_t stream) {
  const float* memory = (const float*)d_in[0];
  const float* cs     = (const float*)d_in[1];
  const float* prev   = (const float*)d_in[2];
  const float* Wk     = (const float*)d_in[3];
  const float* Wb     = (const float*)d_in[4];
  const float* bb     = (const float*)d_in[5];
  const float* Wg     = (const float*)d_in[6];
  const float* bg     = (const float*)d_in[7];
  const float* Ws     = (const float*)d_in[8];
  const float* bs     = (const float*)d_in[9];
  const float* Wgam   = (const float*)d_in[10];
  const float* bgam   = (const float*)d_in[11];
  float* out = (float*)d_out;

  // Workspace layout: z (2 MB) | key (32 KB) | params (2 KB)
  float* zbuf   = (float*)d_ws;
  float* keyb   = zbuf + (size_t)NTM_B * NTM_N;
  float* params = keyb + (size_t)NTM_B * NTM_D;

  // key = cs @ Wk via f32 WMMA (grid: 8 col-tiles x 4 row-tiles, 1 wave each)
  ntm_key_gemm<<<dim3(NTM_D / 16, NTM_B / 16), 32, 0, stream>>>(cs, Wk, keyb);
  // per-batch scalars
  ntm_params<<<NTM_B, 256, 0, stream>>>(cs, keyb, Wb, bb, Wg, bg, Ws, bs,
                                        Wgam, bgam, params);
  // streaming dot + norm + beta scaling (wave per 512B row, NT loads)
  ntm_dot<<<dim3(NTM_N / 64, NTM_B), 256, 0, stream>>>(memory, keyb, params, zbuf);
  // fused softmax / gate / circular conv / sharpen / normalize
  ntm_rowops<<<NTM_B, 1024, 0, stream>>>(zbuf, prev, params, out);
}